// PSRoIPool_16819091931425
// MI455X (gfx1250) — compile-verified
//
#include <hip/hip_runtime.h>
#include <stdint.h>

// PSRoI pooling for MI455X (gfx1250, wave32).
// features: [B, 490, 38, 38] f32 ; rois: [R, 5] f32 ; out: [R, 490] f32
// Latency-bound gather kernel: one block per RoI, 256 threads cover 490
// position-sensitive channels in 2 iterations. CDNA5 paths used:
//   - global_load_async_to_lds_b32 + s_wait_asynccnt (async RoI broadcast)
//   - global_prefetch (via __builtin_prefetch) for the next channel plane
//   - non-temporal stores for the write-once output

namespace {
constexpr int   kP = 7;
constexpr int   kC = 490;
constexpr int   kH = 38;
constexpr int   kW = 38;
constexpr float kScale = 0.0625f;
}

__global__ __launch_bounds__(256) void psroi_pool_kernel(
    const float* __restrict__ feat,
    const float* __restrict__ rois,
    float* __restrict__ out)
{
  __shared__ float s_roi[8];
  const int r   = blockIdx.x;
  const int tid = threadIdx.x;

  // Wave 0 (tid<32 is exactly one wave32) stages this RoI's 5 floats into LDS
  // through the CDNA5 async global->LDS path, then waits on ASYNCcnt.
  if (tid < 32) {
    const int li = (tid < 5) ? tid : 4;  // dup lanes rewrite same word: benign
    const unsigned lds_off = (unsigned)(uintptr_t)(&s_roi[li]);
    const unsigned long long ga =
        (unsigned long long)(uintptr_t)(rois + (size_t)r * 5 + li);
    asm volatile("global_load_async_to_lds_b32 %0, %1, off"
                 :: "v"(lds_off), "v"(ga) : "memory");
    asm volatile("s_wait_asynccnt 0" ::: "memory");
  }
  __syncthreads();

  // Caffe/R-FCN coordinate rounding, replicated in f32 exactly as the
  // reference: round-half-even (jnp.round == rintf), +1 on end, *1/16.
  const int   b  = (int)s_roi[0];
  const float xs = rintf(s_roi[1]) * kScale;
  const float ys = rintf(s_roi[2]) * kScale;
  const float xe = rintf(s_roi[3] + 1.0f) * kScale;
  const float ye = rintf(s_roi[4] + 1.0f) * kScale;
  const float bw = fmaxf(xe - xs, 0.1f) / 7.0f;
  const float bh = fmaxf(ye - ys, 0.1f) / 7.0f;

  const float* __restrict__ fb = feat + (size_t)b * kC * kH * kW;
  float* __restrict__ ob = out + (size_t)r * kC;

  for (int c = tid; c < kC; c += 256) {
    const int q = c % 7;            // pw
    const int p = (c / 7) % 7;      // ph  (c = (d*7+p)*7+q)
    const float fp = (float)p, fq = (float)q;

    // Separate rounded mul+add (no FMA contraction) so floor/ceil boundaries
    // match the reference bit-for-bit, then clip to [0, H] / [0, W].
    float hs = floorf(__fadd_rn(__fmul_rn(fp,        bh), ys));
    float he = ceilf (__fadd_rn(__fmul_rn(fp + 1.0f, bh), ys));
    float ws = floorf(__fadd_rn(__fmul_rn(fq,        bw), xs));
    float we = ceilf (__fadd_rn(__fmul_rn(fq + 1.0f, bw), xs));
    hs = fminf(fmaxf(hs, 0.0f), (float)kH);
    he = fminf(fmaxf(he, 0.0f), (float)kH);
    ws = fminf(fmaxf(ws, 0.0f), (float)kW);
    we = fminf(fmaxf(we, 0.0f), (float)kW);

    const int h0 = (int)hs, h1 = (int)he;
    const int w0 = (int)ws, w1 = (int)we;

    const float* __restrict__ cp = fb + (size_t)c * kH * kW;
    if (c + 256 < kC) {
      // Warm L2/WGP$ for the next channel plane this thread will gather from.
      __builtin_prefetch(fb + (size_t)(c + 256) * kH * kW + h0 * kW + w0, 0, 0);
    }

    float sum = 0.0f;
    for (int h = h0; h < h1; ++h) {
      const float* __restrict__ rp = cp + h * kW;
      float rs = 0.0f;                       // row-partial first (matches
      for (int w = w0; w < w1; ++w) rs += rp[w];  // einsum-over-w then h)
      sum += rs;
    }

    const float cnt = (he - hs) * (we - ws);         // bin area (int-valued)
    const float res = (cnt > 0.0f) ? sum / fmaxf(cnt, 1.0f) : 0.0f;
    __builtin_nontemporal_store(res, ob + c);        // write-once output
  }
}

extern "C" void kernel_launch(void* const* d_in, const int* in_sizes, int n_in,
                              void* d_out, int out_size, void* d_ws, size_t ws_size,
                              hipStream_t stream) {
  const float* feat = (const float*)d_in[0];
  const float* rois = (const float*)d_in[1];
  float* out        = (float*)d_out;
  const int n_roi   = in_sizes[1] / 5;   // 256 in the reference setup
  psroi_pool_kernel<<<dim3(n_roi), dim3(256), 0, stream>>>(feat, rois, out);
  (void)n_in; (void)out_size; (void)d_ws; (void)ws_size;
}